// SAGE_85203561218630
// MI455X (gfx1250) — compile-verified
//
#include <hip/hip_runtime.h>
#include <math.h>

// Problem constants (from reference)
#define N0v 991232
#define N1v 90112
#define N2v 8192
#define E0v 901120
#define E1v 81920
#define IN_F 100
#define HIDv 256
#define OUT_F 47

typedef __attribute__((ext_vector_type(2))) float v2f;
typedef __attribute__((ext_vector_type(8))) float v8f;

// ---------------------------------------------------------------- zero fill
__global__ void __launch_bounds__(256) zero_f32(float* __restrict__ p, long n) {
  long i = (long)blockIdx.x * blockDim.x + threadIdx.x;
  long stride = (long)gridDim.x * blockDim.x;
  for (; i < n; i += stride) p[i] = 0.0f;
}

// -------------------------------------------- edge scatter (mean numerator + counts)
// one wave32 per edge; lanes stride the feature dimension
__global__ void __launch_bounds__(256) scatter_mean_accum(
    const float* __restrict__ feat, const int* __restrict__ src,
    const int* __restrict__ dst, float* __restrict__ agg,
    float* __restrict__ cnt, int E, int D) {
  int tid  = blockIdx.x * blockDim.x + threadIdx.x;
  int wid  = tid >> 5;
  int lane = tid & 31;
  int nw   = (gridDim.x * blockDim.x) >> 5;
  for (int e = wid; e < E; e += nw) {
    int s = src[e];
    int d = dst[e];
    const float* fs = feat + (long)s * D;
    float*       ad = agg  + (long)d * D;
    for (int f = lane; f < D; f += 32)
      atomicAdd(&ad[f], fs[f]);
    if (lane == 0) atomicAdd(&cnt[d], 1.0f);
  }
}

// ---------------------------------------------------- fp32 WMMA K-panel helper
// A: [M x K] row-major (per-row scale ascale), B: [K x N] row-major.
// Layout per CDNA5 ISA 7.12.2:
//   A 16x4 f32: lanes 0-15 -> K=0 (.x), K=1 (.y); lanes 16-31 -> K=2 (.x), K=3 (.y)
//   B 4x16 f32: mirrored (row striped across lanes within a VGPR)
__device__ __forceinline__ v8f mma_panel_f32(
    const float* __restrict__ A, int lda, float ascale,
    const float* __restrict__ B, int ldb, int K,
    int m, int n, bool nvalid, int koff, v8f acc) {
  for (int k = 0; k < K; k += 4) {
    v2f a, b;
    a.x = A[(long)m * lda + (k + koff)]     * ascale;
    a.y = A[(long)m * lda + (k + koff + 1)] * ascale;
    b.x = nvalid ? B[(long)(k + koff)     * ldb + n] : 0.0f;
    b.y = nvalid ? B[(long)(k + koff + 1) * ldb + n] : 0.0f;
    acc = __builtin_amdgcn_wmma_f32_16x16x4_f32(
        /*neg_a=*/false, a, /*neg_b=*/false, b,
        /*c_mod=*/(short)0, acc, /*reuse_a=*/false, /*reuse_b=*/false);
  }
  return acc;
}

// --------------------------------------------------------------- layer-0 GEMM
// h[N1,256] = relu( (agg0/cnt) @ Wl + bl + x[:N1] @ Wr )
__global__ void __launch_bounds__(256) sage_gemm0(
    const float* __restrict__ agg0, const float* __restrict__ cnt0,
    const float* __restrict__ xr,
    const float* __restrict__ Wl, const float* __restrict__ bl,
    const float* __restrict__ Wr, float* __restrict__ h) {
  int w    = (blockIdx.x * blockDim.x + threadIdx.x) >> 5;
  int lane = threadIdx.x & 31;
  int mtile = w >> 4;        // HID/16 = 16 n-tiles
  int ntile = w & 15;
  if (mtile >= N1v / 16) return;

  int m    = mtile * 16 + (lane & 15);
  int n    = ntile * 16 + (lane & 15);
  int koff = (lane >> 4) << 1;

  float ascale = 1.0f / fmaxf(cnt0[m], 1.0f);
  v8f acc = {0.f, 0.f, 0.f, 0.f, 0.f, 0.f, 0.f, 0.f};
  acc = mma_panel_f32(agg0, IN_F, ascale, Wl, HIDv, IN_F, m, n, true, koff, acc);
  acc = mma_panel_f32(xr,   IN_F, 1.0f,   Wr, HIDv, IN_F, m, n, true, koff, acc);

  // C/D layout: lanes 0-15 -> rows r..r ; lanes 16-31 -> rows r+8
  float bias   = bl[n];
  int  rowbase = mtile * 16 + ((lane >> 4) << 3);
#pragma unroll
  for (int r = 0; r < 8; ++r) {
    float v = acc[r] + bias;
    h[(long)(rowbase + r) * HIDv + n] = v > 0.0f ? v : 0.0f;
  }
}

// --------------------------------------------------------------- layer-1 GEMM
// logits[N2,47] = (agg1/cnt) @ Wl + bl + h[:N2] @ Wr   (N padded to 48)
__global__ void __launch_bounds__(256) sage_gemm1(
    const float* __restrict__ agg1, const float* __restrict__ cnt1,
    const float* __restrict__ h,
    const float* __restrict__ Wl, const float* __restrict__ bl,
    const float* __restrict__ Wr, float* __restrict__ logits) {
  const int NT = (OUT_F + 15) / 16;  // 3 column tiles
  int w    = (blockIdx.x * blockDim.x + threadIdx.x) >> 5;
  int lane = threadIdx.x & 31;
  int mtile = w / NT;
  int ntile = w % NT;
  if (mtile >= N2v / 16) return;

  int  m      = mtile * 16 + (lane & 15);
  int  n      = ntile * 16 + (lane & 15);
  bool nvalid = n < OUT_F;
  int  koff   = (lane >> 4) << 1;

  float ascale = 1.0f / fmaxf(cnt1[m], 1.0f);
  v8f acc = {0.f, 0.f, 0.f, 0.f, 0.f, 0.f, 0.f, 0.f};
  acc = mma_panel_f32(agg1, HIDv, ascale, Wl, OUT_F, HIDv, m, n, nvalid, koff, acc);
  acc = mma_panel_f32(h,    HIDv, 1.0f,   Wr, OUT_F, HIDv, m, n, nvalid, koff, acc);

  float bias   = nvalid ? bl[n] : 0.0f;
  int  rowbase = mtile * 16 + ((lane >> 4) << 3);
  if (nvalid) {
#pragma unroll
    for (int r = 0; r < 8; ++r)
      logits[(long)(rowbase + r) * OUT_F + n] = acc[r] + bias;
  }
}

// --------------------------------------------------- row-wise log_softmax (47)
__global__ void __launch_bounds__(256) sage_logsoftmax(float* __restrict__ logits) {
  int row  = blockIdx.x * (blockDim.x >> 5) + (threadIdx.x >> 5);
  int lane = threadIdx.x & 31;
  if (row >= N2v) return;
  float* rp = logits + (long)row * OUT_F;

  float v0 = (lane < OUT_F)      ? rp[lane]      : -3.4e38f;
  float v1 = (lane + 32 < OUT_F) ? rp[lane + 32] : -3.4e38f;

  float mx = fmaxf(v0, v1);
#pragma unroll
  for (int off = 16; off > 0; off >>= 1)
    mx = fmaxf(mx, __shfl_xor(mx, off, 32));

  float s = 0.0f;
  if (lane < OUT_F)      s += expf(v0 - mx);
  if (lane + 32 < OUT_F) s += expf(v1 - mx);
#pragma unroll
  for (int off = 16; off > 0; off >>= 1)
    s += __shfl_xor(s, off, 32);

  float lse = mx + logf(s);
  if (lane < OUT_F)      rp[lane]      = v0 - lse;
  if (lane + 32 < OUT_F) rp[lane + 32] = v1 - lse;
}

// --------------------------------------------------------------------- driver
extern "C" void kernel_launch(void* const* d_in, const int* in_sizes, int n_in,
                              void* d_out, int out_size, void* d_ws, size_t ws_size,
                              hipStream_t stream) {
  const float* x   = (const float*)d_in[0];
  const int*   e0s = (const int*)  d_in[1];
  const int*   e0d = (const int*)  d_in[2];
  const int*   e1s = (const int*)  d_in[3];
  const int*   e1d = (const int*)  d_in[4];
  const float* Wl0 = (const float*)d_in[5];
  const float* bl0 = (const float*)d_in[6];
  const float* Wr0 = (const float*)d_in[7];
  const float* Wl1 = (const float*)d_in[8];
  const float* bl1 = (const float*)d_in[9];
  const float* Wr1 = (const float*)d_in[10];
  float*       out = (float*)d_out;

  // Workspace layout (floats): [agg0 | cnt0 | agg1 | cnt1 | h]
  float* ws   = (float*)d_ws;
  float* agg0 = ws;
  float* cnt0 = agg0 + (long)N1v * IN_F;
  float* agg1 = cnt0 + N1v;
  float* cnt1 = agg1 + (long)N2v * HIDv;
  float* h    = cnt1 + N2v;
  long   zcount = (long)N1v * IN_F + N1v + (long)N2v * HIDv + N2v;

  // 1) zero the atomic accumulators (required each call: graph replay)
  zero_f32<<<4096, 256, 0, stream>>>(ws, zcount);

  // 2) layer-0 edge scatter: one wave per edge, 100 feats
  {
    int blocks = (int)(((long)E0v * 32 + 255) / 256);
    scatter_mean_accum<<<blocks, 256, 0, stream>>>(x, e0s, e0d, agg0, cnt0, E0v, IN_F);
  }

  // 3) layer-0 GEMM + bias + relu -> h[N1,256]
  {
    int tiles = (N1v / 16) * (HIDv / 16);  // 90112 tiles
    sage_gemm0<<<tiles / 8, 256, 0, stream>>>(agg0, cnt0, x, Wl0, bl0, Wr0, h);
  }

  // 4) layer-1 edge scatter: one wave per edge, 256 feats
  {
    int blocks = (int)(((long)E1v * 32 + 255) / 256);
    scatter_mean_accum<<<blocks, 256, 0, stream>>>(h, e1s, e1d, agg1, cnt1, E1v, HIDv);
  }

  // 5) layer-1 GEMM + bias -> logits directly in d_out
  {
    int tiles = (N2v / 16) * ((OUT_F + 15) / 16);  // 1536 tiles
    sage_gemm1<<<tiles / 8, 256, 0, stream>>>(agg1, cnt1, h, Wl1, bl1, Wr1, out);
  }

  // 6) in-place log_softmax over 47 classes, one wave32 per row
  sage_logsoftmax<<<N2v / 8, 256, 0, stream>>>(out);
}